// DecoderLSTM_86328842650330
// MI455X (gfx1250) — compile-verified
//
#include <hip/hip_runtime.h>
#include <math.h>

#define EMB    1024
#define HID    1024
#define VOCAB  50257
#define KSPLIT 4            // K-chunks per LSTM GEMV (parallelism for the 16.8MB matrices)
#define KCH    (HID / KSPLIT)   // 256
#define REDB   64           // partial-reduction blocks for log-softmax

typedef __attribute__((ext_vector_type(2))) float v2f;
typedef __attribute__((ext_vector_type(8))) float v8f;

// fp32 WMMA: C[16,16] += A[16,4] x B[4,16]   (full fp32 precision path)
__device__ __forceinline__ v8f wmma4(v2f a, v2f b, v8f c) {
  return __builtin_amdgcn_wmma_f32_16x16x4_f32(
      /*neg_a=*/false, a, /*neg_b=*/false, b,
      /*c_mod=*/(short)0, c, /*reuse_a=*/false, /*reuse_b=*/false);
}

__device__ __forceinline__ float sigm(float x) { return 1.0f / (1.0f + expf(-x)); }

// ---------------------------------------------------------------- embedding
__global__ void k_embed_relu(const int* __restrict__ word,
                             const float* __restrict__ emb,
                             float* __restrict__ x) {
  int i = blockIdx.x * blockDim.x + threadIdx.x;
  if (i < EMB) {
    float v = emb[(size_t)word[0] * EMB + i];
    x[i] = v > 0.0f ? v : 0.0f;
  }
}

// ---------------------------------------------------------------- LSTM GEMV (partials)
// grid: (64, KSPLIT).  blockIdx.x -> 16 hidden units; blockIdx.y -> K-chunk.
// block: 256 threads = 8 waves. waves 0-3: W_ih gate g; waves 4-7: W_hh gate g.
// Writes combined (ih+hh) partial gate sums for its K-chunk to part[blockIdx.y][4*HID].
__global__ void __launch_bounds__(256) k_lstm_gemv(
    const float* __restrict__ W_ih, const float* __restrict__ W_hh,
    const float* __restrict__ x_in, const float* __restrict__ h_prev,
    float* __restrict__ part) {
  __shared__ __align__(16) float s_x[KCH];
  __shared__ __align__(16) float s_h[KCH];
  __shared__ float s_part[8][16];

  const int kbase = blockIdx.y * KCH;
  if (threadIdx.x < KCH) {
    s_x[threadIdx.x] = x_in[kbase + threadIdx.x];
    s_h[threadIdx.x] = h_prev[kbase + threadIdx.x];
  }
  __syncthreads();

  const int w    = threadIdx.x >> 5;
  const int lane = threadIdx.x & 31;
  const int g    = w & 3;
  const float* Wm  = (w < 4) ? W_ih : W_hh;
  const float* vec = (w < 4) ? s_x  : s_h;
  const int koff = (lane >> 4) << 1;                    // 0 (lanes 0-15), 2 (lanes 16-31)
  const int row  = g * HID + (blockIdx.x << 4) + (lane & 15);
  const float* wrow = Wm + (size_t)row * HID + kbase + koff;

  v8f acc = {};
#pragma unroll 4
  for (int k = 0; k < KCH; k += 4) {
    v2f a = *(const v2f*)(wrow + k);                    // contiguous 8B per lane
    v2f b = *(const v2f*)(vec + k + koff);              // broadcast vector chunk (LDS)
    acc = wmma4(a, b, acc);
  }

  // C layout: lanes 0-15 vgpr r -> M=r ; lanes 16-31 vgpr r -> M=8+r (N columns identical)
  if (lane == 0) {
#pragma unroll
    for (int r = 0; r < 8; ++r) s_part[w][r] = acc[r];
  } else if (lane == 16) {
#pragma unroll
    for (int r = 0; r < 8; ++r) s_part[w][8 + r] = acc[r];
  }
  __syncthreads();

  if (threadIdx.x < 16) {
    int t = threadIdx.x;
    int u = (blockIdx.x << 4) + t;
    float* pp = part + (size_t)blockIdx.y * (4 * HID);
    pp[u]           = s_part[0][t] + s_part[4][t];
    pp[HID + u]     = s_part[1][t] + s_part[5][t];
    pp[2 * HID + u] = s_part[2][t] + s_part[6][t];
    pp[3 * HID + u] = s_part[3][t] + s_part[7][t];
  }
}

// ---------------------------------------------------------------- LSTM elementwise
__global__ void k_lstm_elem(const float* __restrict__ part,
                            const float* __restrict__ b_ih, const float* __restrict__ b_hh,
                            const float* __restrict__ c_prev,
                            float* __restrict__ h_ws, float* __restrict__ h_out,
                            float* __restrict__ c_out) {
  int u = blockIdx.x * blockDim.x + threadIdx.x;
  if (u >= HID) return;
  float gi = 0.f, gf = 0.f, gg = 0.f, go = 0.f;
#pragma unroll
  for (int p = 0; p < KSPLIT; ++p) {
    const float* pp = part + (size_t)p * (4 * HID);
    gi += pp[u];
    gf += pp[HID + u];
    gg += pp[2 * HID + u];
    go += pp[3 * HID + u];
  }
  gi += b_ih[u]           + b_hh[u];
  gf += b_ih[HID + u]     + b_hh[HID + u];
  gg += b_ih[2 * HID + u] + b_hh[2 * HID + u];
  go += b_ih[3 * HID + u] + b_hh[3 * HID + u];
  float cp = c_prev[u];
  float cn = sigm(gf) * cp + sigm(gi) * tanhf(gg);
  float hn = sigm(go) * tanhf(cn);
  h_ws[u]  = hn;
  h_out[u] = hn;
  c_out[u] = cn;
}

// ---------------------------------------------------------------- vocab GEMV
// one wave per 16-row tile of W_out; 8 waves/block -> 128 rows per block
__global__ void __launch_bounds__(256) k_logits(
    const float* __restrict__ W_out, const float* __restrict__ b_out,
    const float* __restrict__ h2, float* __restrict__ logits) {
  __shared__ __align__(16) float s_h[HID];
  for (int i = threadIdx.x; i < HID; i += 256) s_h[i] = h2[i];
  __syncthreads();

  const int w    = threadIdx.x >> 5;
  const int lane = threadIdx.x & 31;
  const int tile = blockIdx.x * 8 + w;
  const int base = tile * 16;
  int row  = base + (lane & 15);
  int rowc = row < VOCAB ? row : VOCAB - 1;             // clamp tail tile (stores guarded)
  const int koff = (lane >> 4) << 1;
  const float* wrow = W_out + (size_t)rowc * HID + koff;

  v8f acc = {};
#pragma unroll 4
  for (int k = 0; k < HID; k += 4) {
    v2f a = *(const v2f*)(wrow + k);
    v2f b = *(const v2f*)(s_h + k + koff);
    acc = wmma4(a, b, acc);
  }

  if (lane == 0) {
#pragma unroll
    for (int r = 0; r < 8; ++r) {
      int rr = base + r;
      if (rr < VOCAB) logits[rr] = acc[r] + b_out[rr];
    }
  } else if (lane == 16) {
#pragma unroll
    for (int r = 0; r < 8; ++r) {
      int rr = base + 8 + r;
      if (rr < VOCAB) logits[rr] = acc[r] + b_out[rr];
    }
  }
}

// ---------------------------------------------------------------- log-softmax (3-stage, parallel)
__global__ void __launch_bounds__(256) k_red_partial(const float* __restrict__ logits,
                                                     float* __restrict__ redM,
                                                     float* __restrict__ redS) {
  __shared__ float red[256];
  const int tid = threadIdx.x;
  const int start = blockIdx.x * 256 + tid;
  const int stride = REDB * 256;

  float m = -INFINITY;
  for (int i = start; i < VOCAB; i += stride) m = fmaxf(m, logits[i]);
  red[tid] = m; __syncthreads();
  for (int s = 128; s > 0; s >>= 1) {
    if (tid < s) red[tid] = fmaxf(red[tid], red[tid + s]);
    __syncthreads();
  }
  float mb = red[0];
  __syncthreads();

  float sum = 0.0f;
  for (int i = start; i < VOCAB; i += stride) sum += expf(logits[i] - mb);
  red[tid] = sum; __syncthreads();
  for (int s = 128; s > 0; s >>= 1) {
    if (tid < s) red[tid] += red[tid + s];
    __syncthreads();
  }
  if (tid == 0) { redM[blockIdx.x] = mb; redS[blockIdx.x] = red[0]; }
}

__global__ void k_red_final(const float* __restrict__ redM,
                            const float* __restrict__ redS,
                            float* __restrict__ logZ) {
  if (threadIdx.x == 0) {
    float M = -INFINITY;
    for (int i = 0; i < REDB; ++i) M = fmaxf(M, redM[i]);
    float S = 0.0f;
    for (int i = 0; i < REDB; ++i) S += redS[i] * expf(redM[i] - M);
    logZ[0] = M + logf(S);
  }
}

__global__ void k_sub(float* __restrict__ out, const float* __restrict__ logZ) {
  int i = blockIdx.x * blockDim.x + threadIdx.x;
  if (i < VOCAB) out[i] -= logZ[0];
}

// ---------------------------------------------------------------- launcher
extern "C" void kernel_launch(void* const* d_in, const int* in_sizes, int n_in,
                              void* d_out_v, int out_size, void* d_ws_v, size_t ws_size,
                              hipStream_t stream) {
  (void)in_sizes; (void)n_in; (void)out_size; (void)ws_size;

  const int*   word  = (const int*)  d_in[0];
  const float* h0    = (const float*)d_in[1];
  const float* c0    = (const float*)d_in[2];
  const float* emb   = (const float*)d_in[3];
  const float* W_ih0 = (const float*)d_in[4];
  const float* W_hh0 = (const float*)d_in[5];
  const float* b_ih0 = (const float*)d_in[6];
  const float* b_hh0 = (const float*)d_in[7];
  const float* W_ih1 = (const float*)d_in[8];
  const float* W_hh1 = (const float*)d_in[9];
  const float* b_ih1 = (const float*)d_in[10];
  const float* b_hh1 = (const float*)d_in[11];
  const float* W_out = (const float*)d_in[12];
  const float* b_out = (const float*)d_in[13];

  float* out = (float*)d_out_v;
  float* ws  = (float*)d_ws_v;
  float* x    = ws;                    // 1024
  float* h1   = ws + 1024;             // 1024
  float* h2   = ws + 2048;             // 1024
  float* part = ws + 4096;             // KSPLIT * 4096 = 16384
  float* redM = ws + 4096 + KSPLIT * 4 * HID;        // 64
  float* redS = redM + REDB;                          // 64
  float* logZ = redS + REDB;                          // 1

  float* out_h1 = out + VOCAB;
  float* out_h2 = out + VOCAB + HID;
  float* out_c1 = out + VOCAB + 2 * HID;
  float* out_c2 = out + VOCAB + 3 * HID;

  k_embed_relu<<<dim3(4), dim3(256), 0, stream>>>(word, emb, x);

  // layer 0
  k_lstm_gemv<<<dim3(64, KSPLIT), dim3(256), 0, stream>>>(W_ih0, W_hh0, x, h0, part);
  k_lstm_elem<<<dim3(4), dim3(256), 0, stream>>>(part, b_ih0, b_hh0, c0,
                                                 h1, out_h1, out_c1);
  // layer 1
  k_lstm_gemv<<<dim3(64, KSPLIT), dim3(256), 0, stream>>>(W_ih1, W_hh1, h1, h0 + HID, part);
  k_lstm_elem<<<dim3(4), dim3(256), 0, stream>>>(part, b_ih1, b_hh1, c0 + HID,
                                                 h2, out_h2, out_c2);

  // vocab projection + log-softmax
  k_logits<<<dim3((VOCAB + 127) / 128), dim3(256), 0, stream>>>(W_out, b_out, h2, out);
  k_red_partial<<<dim3(REDB), dim3(256), 0, stream>>>(out, redM, redS);
  k_red_final<<<dim3(1), dim3(64), 0, stream>>>(redM, redS, logZ);
  k_sub<<<dim3((VOCAB + 255) / 256), dim3(256), 0, stream>>>(out, logZ);
}